// mLSTMCell_45707041964056
// MI455X (gfx1250) — compile-verified
//
#include <hip/hip_runtime.h>
#include <hip/hip_bf16.h>

// mLSTM cell for MI455X (gfx1250, wave32).
// B=128, I=512, H=512.
// d_in order: x[128,512], h_prev[128,512](unused), C_prev[128,512,512],
//             m_prev[128,512], n_prev[128,512], W[3072,512], b[3072]
// d_out: h_t[128,512] | C_t[128,512,512] | m_t[128,512] | n_t[128,512]

typedef __attribute__((ext_vector_type(2))) float v2f;
typedef __attribute__((ext_vector_type(4))) float v4f;
typedef __attribute__((ext_vector_type(8))) float v8f;

#define Bsz 128
#define Hsz 512
#define SIXH 3072
#define INV_SQRT_H 0.04419417382415922f /* 1/sqrt(512) */

// ---------------------------------------------------------------------------
// Kernel 1: gates = x @ W^T + bias   (M=128, N=3072, K=512), fp32 WMMA.
// One 16x16 tile per wave, 4 waves per block.
// A-layout (16x4 f32, ISA 7.12.2): lane<16 holds row=lane, K={k,k+1};
//                                  lane>=16 holds row=lane-16, K={k+2,k+3}.
// D-layout: VGPR r -> rows m0+r (lanes 0-15) and m0+r+8 (lanes 16-31).
// ---------------------------------------------------------------------------
__global__ __launch_bounds__(128) void gates_gemm(
    const float* __restrict__ x,     // [128,512]
    const float* __restrict__ W,     // [3072,512]
    const float* __restrict__ bias,  // [3072]
    float* __restrict__ gates)       // [128,3072]
{
    const int lane = threadIdx.x & 31;
    const int wave = threadIdx.x >> 5;
    const int tile = blockIdx.x * 4 + wave;      // 1536 tiles total
    const int NT_N = SIXH / 16;                  // 192
    const int m0 = (tile / NT_N) * 16;
    const int n0 = (tile % NT_N) * 16;

    const bool hi  = lane >= 16;
    const int  l   = lane & 15;
    const int  koff = hi ? 2 : 0;

    const float* __restrict__ xrow = x + (size_t)(m0 + l) * 512;  // A row for this lane
    const float* __restrict__ wrow = W + (size_t)(n0 + l) * 512;  // B col (= W row)

    v8f acc = {};
#pragma unroll 4
    for (int k = 0; k < 512; k += 4) {
        v2f a, b;
        a.x = xrow[k + koff];
        a.y = xrow[k + koff + 1];
        b.x = wrow[k + koff];
        b.y = wrow[k + koff + 1];
        acc = __builtin_amdgcn_wmma_f32_16x16x4_f32(
            /*neg_a=*/false, a, /*neg_b=*/false, b,
            /*c_mod=*/(short)0, acc, /*reuse_a=*/false, /*reuse_b=*/false);
    }

    const int col = n0 + l;
    const float bv = bias[col];
#pragma unroll
    for (int r = 0; r < 8; ++r) {
        const int row = m0 + r + (hi ? 8 : 0);
        gates[(size_t)row * SIXH + col] = acc[r] + bv;
    }
}

// ---------------------------------------------------------------------------
// Kernel 2: gate nonlinearities + n_t + nq reduction. One block per batch.
// ---------------------------------------------------------------------------
__global__ __launch_bounds__(512) void gate_pointwise(
    const float* __restrict__ gates,   // [128,3072]
    const float* __restrict__ m_prev,  // [128,512]
    const float* __restrict__ n_prev,  // [128,512]
    float* __restrict__ m_out,         // [128,512]
    float* __restrict__ n_out,         // [128,512]
    float* __restrict__ f_ws,          // [128,512]
    float* __restrict__ add_ws,        // [128,512]  i_t*v*k_t
    float* __restrict__ o_ws,          // [128,512]
    float* __restrict__ dinv_ws)       // [128]
{
    const int b = blockIdx.x;
    const int h = threadIdx.x;
    const float* __restrict__ g = gates + (size_t)b * SIXH;

    const float ig = g[h];
    const float fg = g[Hsz + h];
    const float og = g[2 * Hsz + h];
    const float q  = g[3 * Hsz + h];
    const float kk = g[4 * Hsz + h];
    const float v  = g[5 * Hsz + h];

    const float kt = kk * INV_SQRT_H;
    const float mp = m_prev[(size_t)b * Hsz + h];
    const float mt = fmaxf(fg + mp, ig);
    const float it = __expf(ig - mt);
    const float ft = __expf(fg + mp - mt);
    const float ot = 1.0f / (1.0f + __expf(-og));
    const float nt = ft * n_prev[(size_t)b * Hsz + h] + it * kt;

    const size_t idx = (size_t)b * Hsz + h;
    m_out[idx]  = mt;
    n_out[idx]  = nt;
    f_ws[idx]   = ft;
    add_ws[idx] = it * v * kt;
    o_ws[idx]   = ot;

    // nq = dot(n_t, q) over h
    __shared__ float red[Hsz];
    red[h] = nt * q;
    __syncthreads();
#pragma unroll
    for (int s = Hsz / 2; s > 0; s >>= 1) {
        if (h < s) red[h] += red[h + s];
        __syncthreads();
    }
    if (h == 0) {
        const float nq = red[0];
        dinv_ws[b] = 1.0f / fmaxf(fabsf(nq), 1e-6f);
    }
}

// ---------------------------------------------------------------------------
// Kernel 3 (HBM-roofline kernel): C_t = f*C_prev + add, fused Cq = C_t @ q.
// 256 threads = 8 wave32; each wave owns one 512-float C row.
// Non-temporal b128 load/store: C is a single-use 268 MB stream.
// ---------------------------------------------------------------------------
__global__ __launch_bounds__(256) void c_update(
    const float* __restrict__ C_prev,  // [128,512,512]
    const float* __restrict__ gates,   // [128,3072] (q at +3H)
    const float* __restrict__ f_ws,
    const float* __restrict__ add_ws,
    const float* __restrict__ o_ws,
    const float* __restrict__ dinv_ws,
    float* __restrict__ C_out,         // [128,512,512]
    float* __restrict__ h_out)         // [128,512]
{
    const int ROWS_PER_BLOCK = 8;                    // one row per wave
    const int blocksPerBatch = Hsz / ROWS_PER_BLOCK; // 64
    const int b    = blockIdx.x / blocksPerBatch;
    const int rg   = blockIdx.x % blocksPerBatch;
    const int wave = threadIdx.x >> 5;
    const int lane = threadIdx.x & 31;
    const int i    = rg * ROWS_PER_BLOCK + wave;

    __shared__ float qs[Hsz];
    const float* __restrict__ q = gates + (size_t)b * SIXH + 3 * Hsz;
    for (int t = threadIdx.x; t < Hsz; t += 256) qs[t] = q[t];
    __syncthreads();

    const size_t rowIdx = (size_t)b * Hsz + i;
    const float f   = f_ws[rowIdx];
    const float add = add_ws[rowIdx];

    const v4f* __restrict__ src = (const v4f*)(C_prev + rowIdx * Hsz);
    v4f* __restrict__       dst = (v4f*)(C_out + rowIdx * Hsz);

    float acc = 0.0f;
#pragma unroll
    for (int t = lane; t < Hsz / 4; t += 32) {           // 4 iters/lane
        v4f c = __builtin_nontemporal_load(src + t);     // TH=NT b128 load
        c = f * c + add;                                 // splat FMA
        __builtin_nontemporal_store(c, dst + t);         // TH=NT b128 store
        const v4f qv = *(const v4f*)&qs[t * 4];
        acc += c.x * qv.x + c.y * qv.y + c.z * qv.z + c.w * qv.w;
    }

    // wave32 reduction
#pragma unroll
    for (int off = 16; off > 0; off >>= 1)
        acc += __shfl_down(acc, off, 32);

    if (lane == 0)
        h_out[rowIdx] = o_ws[rowIdx] * acc * dinv_ws[b];
}

// ---------------------------------------------------------------------------
extern "C" void kernel_launch(void* const* d_in, const int* in_sizes, int n_in,
                              void* d_out, int out_size, void* d_ws, size_t ws_size,
                              hipStream_t stream) {
    (void)in_sizes; (void)n_in; (void)out_size; (void)ws_size;

    const float* x      = (const float*)d_in[0];
    // d_in[1] = h_prev (unused by the recurrence outputs)
    const float* C_prev = (const float*)d_in[2];
    const float* m_prev = (const float*)d_in[3];
    const float* n_prev = (const float*)d_in[4];
    const float* W      = (const float*)d_in[5];
    const float* bias   = (const float*)d_in[6];

    // d_out layout: h_t | C_t | m_t | n_t
    float* h_out = (float*)d_out;
    float* C_out = h_out + (size_t)Bsz * Hsz;
    float* m_out = C_out + (size_t)Bsz * Hsz * Hsz;
    float* n_out = m_out + (size_t)Bsz * Hsz;

    // Workspace layout (floats)
    float* ws     = (float*)d_ws;
    float* gates  = ws;                                   // 128*3072
    float* f_ws   = gates + (size_t)Bsz * SIXH;           // 128*512
    float* add_ws = f_ws + (size_t)Bsz * Hsz;
    float* o_ws   = add_ws + (size_t)Bsz * Hsz;
    float* dinv   = o_ws + (size_t)Bsz * Hsz;             // 128

    // 1) gate GEMM: 1536 tiles, 4 waves/block -> 384 blocks
    gates_gemm<<<384, 128, 0, stream>>>(x, W, bias, gates);

    // 2) pointwise gates + nq reduction: one block per batch
    gate_pointwise<<<Bsz, Hsz, 0, stream>>>(gates, m_prev, n_prev,
                                            m_out, n_out, f_ws, add_ws, o_ws, dinv);

    // 3) streaming C update + fused readout: 128 * 64 = 8192 blocks
    c_update<<<Bsz * (Hsz / 8), 256, 0, stream>>>(C_prev, gates, f_ws, add_ws,
                                                  o_ws, dinv, C_out, h_out);
}